// GCN_12841952215390
// MI455X (gfx1250) — compile-verified
//
#include <hip/hip_runtime.h>
#include <hip/hip_bf16.h>
#include <stdint.h>

#define IN_F 32
#define HID  128

typedef __attribute__((ext_vector_type(2))) float v2f;
typedef __attribute__((ext_vector_type(8))) float v8f;

// ---------------------------------------------------------------------------
// Degree / normalization precompute
// ---------------------------------------------------------------------------
__global__ void init_deg_kernel(float* __restrict__ deg, int n) {
    long long i = (long long)blockIdx.x * blockDim.x + threadIdx.x;
    if (i < n) deg[i] = 1.0f;  // self-loop contributes 1
}

__global__ void accum_deg_kernel(const long long* __restrict__ dst,
                                 float* __restrict__ deg, long long e) {
    long long i = (long long)blockIdx.x * blockDim.x + threadIdx.x;
    if (i < e) atomicAdd(&deg[dst[i]], 1.0f);
}

__global__ void finalize_dis_kernel(float* __restrict__ deg, int n) {
    long long i = (long long)blockIdx.x * blockDim.x + threadIdx.x;
    if (i < n) {
        float d = deg[i];
        deg[i] = (d > 0.0f) ? rsqrtf(d) : 0.0f;  // in-place: deg -> deg^{-1/2}
    }
}

__global__ void compute_norm_kernel(const long long* __restrict__ src,
                                    const long long* __restrict__ dst,
                                    const float* __restrict__ dis,
                                    float* __restrict__ norm, long long e) {
    long long i = (long long)blockIdx.x * blockDim.x + threadIdx.x;
    if (i < e) norm[i] = dis[src[i]] * dis[dst[i]];
}

// ---------------------------------------------------------------------------
// Dense transform H = X @ W via V_WMMA_F32_16X16X4_F32.
//
// Requires nNodes % 16 == 0 (true here: 100000).
// Block = 256 threads = 8 waves. W is staged into LDS once per block in a
// K-PAIR SWIZZLED layout: sWp[kp*OUT*2 + c*2 + {0,1}] = W[2kp][c], W[2kp+1][c]
// so every B fragment (v2f: K and K+1 of one column) is ONE contiguous,
// 8-byte-aligned LDS read -> single ds_load_b64 into an adjacent VGPR pair,
// consumed directly by the WMMA (no register re-pairing moves).
//
// Each wave owns one 16-row strip of X and accumulates the FULL output
// row-strip (OUT/16 tiles) while looping K once:
//   - A frag: single global b64 load per K-step (lane<16: K={0,1},
//     lane>=16: K={2,3} of row lane&15 — per ISA 16x4 f32 layout)
//   - B frags: ds_load_b64 per tile, immediate offsets
// C/D: vgpr i -> M = i + (lane>=16 ? 8 : 0), N = lane & 15.
// ---------------------------------------------------------------------------
template <int IN, int OUT>
__global__ __launch_bounds__(256) void transform_wmma_kernel(
    const float* __restrict__ X, const float* __restrict__ W,
    float* __restrict__ H, int nNodes) {
    constexpr int TN = OUT / 16;
    __shared__ float sWp[IN * OUT];   // k-pair swizzled copy of W (<=64KB)

    // Stage W -> LDS with K-pair swizzle. Global reads fully coalesced.
    for (int i = threadIdx.x; i < IN * OUT; i += 256) {
        int k = i / OUT;
        int c = i - k * OUT;
        sWp[(k >> 1) * (OUT * 2) + c * 2 + (k & 1)] = W[i];
    }
    __syncthreads();

    const int lane   = threadIdx.x & 31;
    const int wave   = threadIdx.x >> 5;
    const int tilesM = nNodes >> 4;
    int tm = blockIdx.x * 8 + wave;
    const bool active = (tm < tilesM);     // wave-uniform
    if (!active) tm = tilesM - 1;          // clamp: stay converged, skip stores

    const int halfK = (lane >> 4) * 2;     // 0 or 2
    const int col   = lane & 15;
    const float* aPtr = X + ((long long)tm * 16 + col) * IN + halfK;
    const float* wCol = sWp + col * 2 + (halfK >> 1) * (OUT * 2);

    v8f acc[TN];
#pragma unroll
    for (int t = 0; t < TN; ++t) acc[t] = (v8f){};

    for (int k0 = 0; k0 < IN; k0 += 4) {
        v2f a = *(const v2f*)(aPtr + k0);                    // one global b64
        const float* wb = wCol + (k0 >> 1) * (OUT * 2);      // LDS base, this K-pair
#pragma unroll
        for (int t = 0; t < TN; ++t) {
            v2f b = *(const v2f*)(wb + t * 32);              // one ds_load_b64
            acc[t] = __builtin_amdgcn_wmma_f32_16x16x4_f32(
                false, a, false, b, (short)0, acc[t], false, false);
        }
    }

    if (active) {
        const int mBase = tm * 16 + ((lane >= 16) ? 8 : 0);
#pragma unroll
        for (int t = 0; t < TN; ++t) {
            float* hp = H + (long long)mBase * OUT + t * 16 + col;
#pragma unroll
            for (int i = 0; i < 8; ++i) hp[(long long)i * OUT] = acc[t][i];
        }
    }
}

// ---------------------------------------------------------------------------
// OUTP[i][c] = b[c] + H[i][c] * dis[i]^2   (bias + self-loop message)
// ---------------------------------------------------------------------------
template <int OUT>
__global__ void init_out_kernel(const float* __restrict__ H,
                                const float* __restrict__ dis,
                                const float* __restrict__ bias,
                                float* __restrict__ OUTP, int nNodes) {
    long long t = (long long)blockIdx.x * blockDim.x + threadIdx.x;
    long long total = (long long)nNodes * OUT;
    if (t >= total) return;
    long long i = t / OUT;
    int c = (int)(t % OUT);
    float s = dis[i];
    OUTP[t] = bias[c] + H[t] * (s * s);
}

// ---------------------------------------------------------------------------
// Edge scatter: OUTP[dst] += H[src] * norm[e].  OUT/4 consecutive threads
// cover one edge (float4 per thread), so the H[src] row gather coalesces.
// ---------------------------------------------------------------------------
template <int OUT>
__global__ void scatter_kernel(const float* __restrict__ H,
                               const long long* __restrict__ src,
                               const long long* __restrict__ dst,
                               const float* __restrict__ norm,
                               float* __restrict__ OUTP, long long nEdges) {
    constexpr int chunks = OUT / 4;
    long long t = (long long)blockIdx.x * blockDim.x + threadIdx.x;
    long long e = t / chunks;
    if (e >= nEdges) return;
    int c = (int)(t % chunks) * 4;
    long long s = src[e];
    long long d = dst[e];
    float w = norm[e];
    const float4 v = *(const float4*)(H + s * OUT + c);
    float* op = OUTP + d * OUT + c;
    atomicAdd(op + 0, v.x * w);
    atomicAdd(op + 1, v.y * w);
    atomicAdd(op + 2, v.z * w);
    atomicAdd(op + 3, v.w * w);
}

__global__ void relu_kernel(float* __restrict__ p, long long n) {
    long long t = (long long)blockIdx.x * blockDim.x + threadIdx.x;
    if (t < n) p[t] = fmaxf(p[t], 0.0f);
}

// ---------------------------------------------------------------------------
// Launch
// ---------------------------------------------------------------------------
static inline unsigned blocks_for(long long work, int tpb) {
    return (unsigned)((work + tpb - 1) / tpb);
}

extern "C" void kernel_launch(void* const* d_in, const int* in_sizes, int n_in,
                              void* d_out, int out_size, void* d_ws, size_t ws_size,
                              hipStream_t stream) {
    const float*      x   = (const float*)d_in[0];
    const long long*  ei  = (const long long*)d_in[1];
    const float*      W1  = (const float*)d_in[2];
    const float*      b1  = (const float*)d_in[3];
    const float*      W2  = (const float*)d_in[4];
    const float*      b2  = (const float*)d_in[5];
    const float*      W3  = (const float*)d_in[6];
    const float*      b3  = (const float*)d_in[7];

    const int       N = in_sizes[0] / IN_F;
    const long long E = (long long)in_sizes[1] / 2;
    const long long* src = ei;
    const long long* dst = ei + E;

    // Workspace layout
    float* H   = (float*)d_ws;                      // N * HID
    float* P   = H + (size_t)N * HID;               // N * HID
    float* dis = P + (size_t)N * HID;               // N   (deg -> deg^-1/2)
    float* nrm = dis + N;                           // E
    float* out = (float*)d_out;                     // N * IN_F

    const int TPB = 256;

    // --- normalization precompute ---
    init_deg_kernel<<<blocks_for(N, TPB), TPB, 0, stream>>>(dis, N);
    accum_deg_kernel<<<blocks_for(E, TPB), TPB, 0, stream>>>(dst, dis, E);
    finalize_dis_kernel<<<blocks_for(N, TPB), TPB, 0, stream>>>(dis, N);
    compute_norm_kernel<<<blocks_for(E, TPB), TPB, 0, stream>>>(src, dst, dis, nrm, E);

    const long long tilesM = N / 16;   // N % 16 == 0
    const unsigned  gemmBlocks = blocks_for(tilesM, 8);

    // --- layer 1: x(N,32) @ W1(32,128) -> P ---
    transform_wmma_kernel<IN_F, HID><<<gemmBlocks, TPB, 0, stream>>>(x, W1, H, N);
    init_out_kernel<HID><<<blocks_for((long long)N * HID, TPB), TPB, 0, stream>>>(H, dis, b1, P, N);
    scatter_kernel<HID><<<blocks_for(E * (HID / 4), TPB), TPB, 0, stream>>>(H, src, dst, nrm, P, E);
    relu_kernel<<<blocks_for((long long)N * HID, TPB), TPB, 0, stream>>>(P, (long long)N * HID);

    // --- layer 2: P(N,128) @ W2(128,128) -> P (P read into H before P rewritten) ---
    transform_wmma_kernel<HID, HID><<<gemmBlocks, TPB, 0, stream>>>(P, W2, H, N);
    init_out_kernel<HID><<<blocks_for((long long)N * HID, TPB), TPB, 0, stream>>>(H, dis, b2, P, N);
    scatter_kernel<HID><<<blocks_for(E * (HID / 4), TPB), TPB, 0, stream>>>(H, src, dst, nrm, P, E);
    relu_kernel<<<blocks_for((long long)N * HID, TPB), TPB, 0, stream>>>(P, (long long)N * HID);

    // --- layer 3: P(N,128) @ W3(128,32) -> d_out ---
    transform_wmma_kernel<HID, IN_F><<<gemmBlocks, TPB, 0, stream>>>(P, W3, H, N);
    init_out_kernel<IN_F><<<blocks_for((long long)N * IN_F, TPB), TPB, 0, stream>>>(H, dis, b3, out, N);
    scatter_kernel<IN_F><<<blocks_for(E * (IN_F / 4), TPB), TPB, 0, stream>>>(H, src, dst, nrm, out, E);
    relu_kernel<<<blocks_for((long long)N * IN_F, TPB), TPB, 0, stream>>>(out, (long long)N * IN_F);
}